// AttnVLADLayer_64269890617648
// MI455X (gfx1250) — compile-verified
//
#include <hip/hip_runtime.h>
#include <math.h>

typedef float v2f __attribute__((ext_vector_type(2)));
typedef float v8f __attribute__((ext_vector_type(8)));
typedef int   v4i __attribute__((vector_size(4 * sizeof(int))));

#define B_   32
#define D_   512
#define N_   4096
#define KC   64
#define NT   256            // n columns per workgroup chunk
#define DS   64             // d rows per LDS slice
#define NCH  (N_ / NT)      // 16 chunks per batch

// LDS pitches (floats): xs/qs rows 16B-aligned, Sm rows 8B, bank-spread for row-varying reads.
#define XS_PITCH 260        // 1040 B rows, bank step 4
#define QS_PITCH 68         // 272 B rows, bank step 4
#define S_PITCH  258        // 1032 B rows, bank step 2

// workspace layout (float offsets)
#define QT_OFF    0                       // qT [KC][D_]        32768 floats
#define DEN_OFF   32768                   // denom [B_][KC]      2048 floats
#define SSQ_OFF   34816                   // ssq [B_]              32 floats
#define DACC_OFF  34880                   // dacc [B_][D_][KC] 1048576 floats
#define ACC_FLOATS (2112 + 1048576)       // zeroed region from DEN_OFF

// device-pass-only sanity checks (host pass has no amdgcn builtins)
#if defined(__HIP_DEVICE_COMPILE__)
#if !__has_builtin(__builtin_amdgcn_global_load_async_to_lds_b128)
#error "no __builtin_amdgcn_global_load_async_to_lds_b128 on this toolchain"
#endif
#if !__has_builtin(__builtin_amdgcn_s_wait_asynccnt)
#error "no __builtin_amdgcn_s_wait_asynccnt on this toolchain"
#endif
#endif

#define AS1 __attribute__((address_space(1)))
#define AS3 __attribute__((address_space(3)))

// async 16B global->LDS copy, tracked by ASYNCcnt
__device__ __forceinline__ void async_cp16(const float* g, float* l) {
  __builtin_amdgcn_global_load_async_to_lds_b128(
      (AS1 v4i*)g, (AS3 v4i*)l, 0, 0);
}

// D = A(16x4 f32) * B(4x16 f32) + C ; wave32, EXEC all ones
__device__ __forceinline__ v8f wmma_f32(v2f a, v2f b, v8f c) {
  return __builtin_amdgcn_wmma_f32_16x16x4_f32(
      /*neg_a=*/false, a, /*neg_b=*/false, b,
      /*c_mod=*/(short)0, c, /*reuse_a=*/false, /*reuse_b=*/false);
}

// Issue async staging of a [DS x NT] x-slice (rows s*DS.., cols n0..) into LDS.
__device__ __forceinline__ void issue_xs_async(float* xs, const float* xb,
                                               int s, int n0, int t) {
#pragma unroll
  for (int i = 0; i < 16; ++i) {
    int fid = t + i * 256;                 // 0..4095 float4 tiles
    int r   = fid >> 6;                    // 0..63
    int c4  = fid & 63;                    // 0..63
    async_cp16(xb + (s * DS + r) * N_ + n0 + c4 * 4,
               xs + r * XS_PITCH + c4 * 4);
  }
}

// Issue async staging of a [KC x DS] qT-slice into LDS.
__device__ __forceinline__ void issue_qs_async(float* qs, const float* qT,
                                               int s, int t) {
#pragma unroll
  for (int i = 0; i < 4; ++i) {
    int fid = t + i * 256;                 // 0..1023 float4 tiles
    int r   = fid >> 4;                    // 0..63 (K row)
    int c4  = fid & 15;
    async_cp16(qT + r * D_ + s * DS + c4 * 4,
               qs + r * QS_PITCH + c4 * 4);
  }
}

// ---------------------------------------------------------------------------
// Kernel 1: qT[k][d] = centers[d][k] / max(||centers[:,k]||, 1e-12)
// ---------------------------------------------------------------------------
__global__ __launch_bounds__(256) void qnorm_kernel(
    const float* __restrict__ centers, float* __restrict__ qT) {
  __shared__ float red[256];
  __shared__ float invs;
  const int k = blockIdx.x;
  const int t = threadIdx.x;
  float s = 0.f;
  for (int dd = t; dd < D_; dd += 256) {
    float c = centers[dd * KC + k];
    s += c * c;
  }
  red[t] = s;
  __syncthreads();
  for (int off = 128; off > 0; off >>= 1) {
    if (t < off) red[t] += red[t + off];
    __syncthreads();
  }
  if (t == 0) invs = 1.0f / fmaxf(sqrtf(red[0]), 1e-12f);
  __syncthreads();
  const float iv = invs;
  for (int dd = t; dd < D_; dd += 256)
    qT[k * D_ + dd] = centers[dd * KC + k] * iv;
}

// ---------------------------------------------------------------------------
// Kernel 2: fused score GEMM + softmax(K) + denom + desc-partial GEMM
// grid = B_ * NCH blocks, 256 threads (8 waves), double-buffered async staging
// ---------------------------------------------------------------------------
__global__ __launch_bounds__(256) void attnvlad_main(
    const float* __restrict__ x, const float* __restrict__ alpha_p,
    const float* __restrict__ qT, float* __restrict__ denom_g,
    float* __restrict__ dacc_g) {
  __shared__ __align__(16) float xs[2][DS * XS_PITCH];  // x slices [64][260]
  __shared__ __align__(16) float qs[2][KC * QS_PITCH];  // qT slices [64][68]
  __shared__ __align__(16) float Sm[KC * S_PITCH];      // score/prob [64][258]
  __shared__ float part[4][KC];

  const int bb = blockIdx.x / NCH;
  const int ch = blockIdx.x % NCH;
  const int n0 = ch * NT;
  const int t    = threadIdx.x;
  const int wave = t >> 5;
  const int lane = t & 31;
  const int lm   = lane & 15;              // lane % 16
  const int lh   = lane >> 4;              // lane half

  const float* xb = x + (size_t)bb * ((size_t)D_ * N_);

  // ---- Phase 1: score[64 x 256] = qT[64 x 512] * x_chunk[512 x 256] ----
  // wave w owns N-tiles {2w, 2w+1}, all 4 M-tiles; accumulate over 8 d-slices
  v8f acc[2][4];
#pragma unroll
  for (int i = 0; i < 2; ++i)
#pragma unroll
    for (int m = 0; m < 4; ++m) acc[i][m] = (v8f)0.0f;

  // prime the pipeline: prefetch slice 0 into buffer 0
  issue_xs_async(xs[0], xb, 0, n0, t);
  issue_qs_async(qs[0], qT, 0, t);
  __builtin_amdgcn_s_wait_asynccnt(0);
  __syncthreads();

  int buf = 0;
  for (int s = 0; s < 8; ++s) {
    if (s < 7) {                            // prefetch next slice (other buffer)
      issue_xs_async(xs[buf ^ 1], xb, s + 1, n0, t);
      issue_qs_async(qs[buf ^ 1], qT, s + 1, t);
    }
    const float* xsb = xs[buf];
    const float* qsb = qs[buf];
#pragma unroll 4
    for (int kk = 0; kk < 16; ++kk) {       // 16 k-steps of 4 within slice
      const int cb = 4 * kk + 2 * lh;
      v2f a[4], b[2];
#pragma unroll
      for (int mt = 0; mt < 4; ++mt)
        a[mt] = *(const v2f*)&qsb[(mt * 16 + lm) * QS_PITCH + cb];
#pragma unroll
      for (int i = 0; i < 2; ++i) {
        const int nt = 2 * wave + i;
        b[i].x = xsb[(cb + 0) * XS_PITCH + nt * 16 + lm];
        b[i].y = xsb[(cb + 1) * XS_PITCH + nt * 16 + lm];
      }
#pragma unroll
      for (int i = 0; i < 2; ++i)
#pragma unroll
        for (int mt = 0; mt < 4; ++mt)
          acc[i][mt] = wmma_f32(a[mt], b[i], acc[i][mt]);
    }
    __builtin_amdgcn_s_wait_asynccnt(0);    // my prefetch landed
    __syncthreads();                        // everyone's landed, readers done
    buf ^= 1;
  }

  // overlap phase-4's first x prefetch with softmax/denom (targets buffer 0,
  // last read at slice 6 which is already fenced by the barriers above)
  issue_xs_async(xs[0], xb, 0, n0, t);

  // spill score tiles to LDS: Sm[kcluster][ncol]
#pragma unroll
  for (int i = 0; i < 2; ++i) {
    const int nt = 2 * wave + i;
#pragma unroll
    for (int mt = 0; mt < 4; ++mt)
#pragma unroll
      for (int v = 0; v < 8; ++v)
        Sm[(mt * 16 + v + 8 * lh) * S_PITCH + nt * 16 + lm] = acc[i][mt][v];
  }
  __syncthreads();

  // ---- Phase 2: softmax over K per column (thread t = column t) ----
  {
    const float alpha = alpha_p[0];
    float mx = -INFINITY;
    for (int k = 0; k < KC; ++k)
      mx = fmaxf(mx, alpha * Sm[k * S_PITCH + t]);
    float sum = 0.f;
    for (int k = 0; k < KC; ++k) {
      float e = __expf(alpha * Sm[k * S_PITCH + t] - mx);
      Sm[k * S_PITCH + t] = e;
      sum += e;
    }
    const float inv = 1.0f / sum;           // sum >= 1, safe
    for (int k = 0; k < KC; ++k) Sm[k * S_PITCH + t] *= inv;
  }
  __syncthreads();

  // ---- Phase 3: partial denom[k] = sum over this chunk's columns ----
  {
    const int k = t & 63, g = t >> 6;
    float p = 0.f;
    for (int c = g * 64; c < g * 64 + 64; ++c) p += Sm[k * S_PITCH + c];
    part[g][k] = p;
  }
  __syncthreads();
  if (t < 64) {
    float ds = part[0][t] + part[1][t] + part[2][t] + part[3][t];
    atomicAdd(denom_g + bb * KC + t, ds);
  }
  __builtin_amdgcn_s_wait_asynccnt(0);      // phase-4 slice 0 landed
  __syncthreads();

  // ---- Phase 4: dacc[512 x 64] += x_chunk[512 x 256] * probT[256 x 64] ----
  const int mtl  = wave >> 1;               // d-tile within slice (0..3)
  const int ntc0 = (wave & 1) * 2;          // first of 2 K-cluster tiles
  buf = 0;
  for (int s = 0; s < 8; ++s) {
    if (s < 7)                              // prefetch next slice
      issue_xs_async(xs[buf ^ 1], xb, s + 1, n0, t);

    const float* xsb = xs[buf];
    v8f d0 = (v8f)0.0f, d1 = (v8f)0.0f;
#pragma unroll 4
    for (int kk = 0; kk < 64; ++kk) {       // 256-wide n contraction
      const int cb = 4 * kk + 2 * lh;
      v2f a  = *(const v2f*)&xsb[(mtl * 16 + lm) * XS_PITCH + cb];
      v2f b0 = *(const v2f*)&Sm[(ntc0 * 16 + lm) * S_PITCH + cb];
      v2f b1 = *(const v2f*)&Sm[((ntc0 + 1) * 16 + lm) * S_PITCH + cb];
      d0 = wmma_f32(a, b0, d0);
      d1 = wmma_f32(a, b1, d1);
    }
    const int drow = s * DS + mtl * 16;
#pragma unroll
    for (int v = 0; v < 8; ++v) {
      const int dd = drow + v + 8 * lh;
      float* base = dacc_g + (size_t)(bb * D_ + dd) * KC;
      atomicAdd(base + ntc0 * 16 + lm,       d0[v]);
      atomicAdd(base + (ntc0 + 1) * 16 + lm, d1[v]);
    }
    __builtin_amdgcn_s_wait_asynccnt(0);
    __syncthreads();
    buf ^= 1;
  }
}

// ---------------------------------------------------------------------------
// Kernel 3: residual + intra-normalize over d, apply cluster weights,
//           accumulate per-batch sum of squares. One block per batch.
// ---------------------------------------------------------------------------
__global__ __launch_bounds__(256) void finalize1_kernel(
    const float* __restrict__ dacc, const float* __restrict__ denom,
    const float* __restrict__ centers, const float* __restrict__ cw,
    float* __restrict__ out, float* __restrict__ ssq_g) {
  __shared__ float red[4][KC];
  __shared__ float invn[KC];
  const int b = blockIdx.x;
  const int t = threadIdx.x;
  const int k = t & 63, g = t >> 6;
  const float invd = 1.0f / fmaxf(denom[b * KC + k], 1e-6f);

  float ss = 0.f;
  for (int dd = g * 128; dd < g * 128 + 128; ++dd) {
    float y = dacc[(size_t)(b * D_ + dd) * KC + k] * invd - centers[dd * KC + k];
    ss += y * y;
  }
  red[g][k] = ss;
  __syncthreads();
  if (t < 64) {
    float tot = red[0][t] + red[1][t] + red[2][t] + red[3][t];
    float iv  = 1.0f / fmaxf(sqrtf(tot), 1e-12f);
    invn[t] = iv;
    float w = cw[t];
    atomicAdd(ssq_g + b, tot * iv * iv * w * w);   // = sum of (y*iv*w)^2
  }
  __syncthreads();
  const float scale = invn[k] * cw[k];
  for (int dd = g * 128; dd < g * 128 + 128; ++dd) {
    float y = dacc[(size_t)(b * D_ + dd) * KC + k] * invd - centers[dd * KC + k];
    out[(size_t)b * (D_ * KC) + dd * KC + k] = y * scale;
  }
}

// ---------------------------------------------------------------------------
// Kernel 4: global L2 normalize per batch
// ---------------------------------------------------------------------------
__global__ __launch_bounds__(256) void finalize2_kernel(
    float* __restrict__ out, const float* __restrict__ ssq_g) {
  const size_t i = (size_t)blockIdx.x * 256 + threadIdx.x;
  if (i < (size_t)B_ * D_ * KC) {
    const int b = (int)(i >> 15);          // D_*KC = 32768
    out[i] *= 1.0f / fmaxf(sqrtf(ssq_g[b]), 1e-12f);
  }
}

extern "C" void kernel_launch(void* const* d_in, const int* in_sizes, int n_in,
                              void* d_out, int out_size, void* d_ws, size_t ws_size,
                              hipStream_t stream) {
  (void)in_sizes; (void)n_in; (void)out_size; (void)ws_size;
  const float* x       = (const float*)d_in[0];   // [32,512,4096]
  const float* centers = (const float*)d_in[1];   // [1,512,64]
  const float* alpha   = (const float*)d_in[2];   // scalar
  const float* cw      = (const float*)d_in[3];   // [1,1,64]
  float* out = (float*)d_out;                     // [32, 32768]
  float* ws  = (float*)d_ws;

  float* qT    = ws + QT_OFF;
  float* denom = ws + DEN_OFF;
  float* ssq   = ws + SSQ_OFF;
  float* dacc  = ws + DACC_OFF;

  // zero accumulators (denom, ssq, dacc) — graph-capturable memset node
  (void)hipMemsetAsync(denom, 0, (size_t)ACC_FLOATS * sizeof(float), stream);

  qnorm_kernel<<<KC, 256, 0, stream>>>(centers, qT);
  attnvlad_main<<<B_ * NCH, 256, 0, stream>>>(x, alpha, qT, denom, dacc);
  finalize1_kernel<<<B_, 256, 0, stream>>>(dacc, denom, centers, cw, out, ssq);
  finalize2_kernel<<<(B_ * D_ * KC) / 256, 256, 0, stream>>>(out, ssq);
}